// AttentionEnhancedBiLSTM_68624987455930
// MI455X (gfx1250) — compile-verified
//
#include <hip/hip_runtime.h>
#include <math.h>

typedef float v2f __attribute__((ext_vector_type(2)));
typedef float v4f __attribute__((ext_vector_type(4)));
typedef float v8f __attribute__((ext_vector_type(8)));

#define B_SZ 1024
#define T_SZ 128
#define E_SZ 1024
#define H_SZ 512

#define BM 128
#define BN 128
#define BK 16
#define S20 20      // padded LDS stride (floats) for m/n-major tiles
#define S_NN 136    // padded LDS k-major row stride (floats) for the NN B tile

__device__ __forceinline__ float sigm(float x) { return 1.0f / (1.0f + __expf(-x)); }

// ---------------------------------------------------------------------------
// CDNA5 async global -> LDS copy, 16 bytes per lane (ASYNCcnt tracked).
// SADDR form: vdst = per-lane LDS byte offset, vaddr = per-lane byte offset,
// saddr = uniform 64-bit base.
// ---------------------------------------------------------------------------
__device__ __forceinline__ void async_ld16(unsigned lds_off, const float* base,
                                           unsigned gbyte_off) {
    asm volatile("global_load_async_to_lds_b128 %0, %1, %2"
                 :: "v"(lds_off), "v"(gbyte_off), "s"(base)
                 : "memory");
}
__device__ __forceinline__ void wait_async0() {
    asm volatile("s_wait_asynccnt 0x0" ::: "memory");
}

// ---------------------------------------------------------------------------
// Extract x = inputs[:, T-1, :] and the feature-reversed xr
// ---------------------------------------------------------------------------
__global__ __launch_bounds__(256) void prep_kernel(const float* __restrict__ inp,
                                                   float* __restrict__ x,
                                                   float* __restrict__ xr) {
    int idx = blockIdx.x * 256 + threadIdx.x;      // over B*E
    int b = idx >> 10;                             // /E (E=1024)
    int e = idx & (E_SZ - 1);
    float v = inp[(size_t)b * T_SZ * E_SZ + (size_t)(T_SZ - 1) * E_SZ + e];
    x[idx] = v;
    xr[b * E_SZ + (E_SZ - 1 - e)] = v;
}

// ---------------------------------------------------------------------------
// Issue async copies for one K-stage (A tile 128x16, B tile 16 wide).
//   BT=true : B is N x K row-major -> n-major LDS tile, stride S20
//   BT=false: B is K x N row-major -> k-major LDS tile, stride S_NN
// 128 threads; 512 16B chunks per tile -> 4 chunks/thread/tile.
// ---------------------------------------------------------------------------
template<bool BT>
__device__ __forceinline__ void stage_tiles(const float* A, int lda,
                                            const float* Bm, int ldb,
                                            int blockM, int blockN, int k0,
                                            unsigned ldsa, unsigned ldsb, int tid) {
    #pragma unroll
    for (int i = 0; i < 4; ++i) {
        int c = tid + i * 128;
        int r = c >> 2, q = c & 3;
        async_ld16(ldsa + (unsigned)(r * S20 + q * 4) * 4u, A,
                   (unsigned)(((blockM + r) * lda + k0 + q * 4) * 4));
    }
    if (BT) {
        #pragma unroll
        for (int i = 0; i < 4; ++i) {
            int c = tid + i * 128;
            int n = c >> 2, q = c & 3;
            async_ld16(ldsb + (unsigned)(n * S20 + q * 4) * 4u, Bm,
                       (unsigned)(((blockN + n) * ldb + k0 + q * 4) * 4));
        }
    } else {
        #pragma unroll
        for (int i = 0; i < 4; ++i) {
            int c = tid + i * 128;
            int kr = c >> 5, cq = c & 31;
            async_ld16(ldsb + (unsigned)(kr * S_NN + cq * 4) * 4u, Bm,
                       (unsigned)(((k0 + kr) * ldb + blockN + cq * 4) * 4));
        }
    }
}

// ---------------------------------------------------------------------------
// Tiled fp32 WMMA GEMM:  C[M x N] = A[M x K] * B' + bias0 + bias1 + resid
//   BT=true : C = A @ B^T (B stored N x K)      BT=false: C = A @ B (K x N)
// 128 threads = 4 waves; block tile 128x128; wave tile 64x64 (4x4 frags).
// Double-buffered LDS filled by GLOBAL_LOAD_ASYNC_TO_LDS_B128.
// ---------------------------------------------------------------------------
template<bool BT>
__global__ __launch_bounds__(128) void gemm_wmma_kernel(
    const float* __restrict__ A, int lda,
    const float* __restrict__ Bm, int ldb,
    const float* __restrict__ bias0,
    const float* __restrict__ bias1,
    const float* __restrict__ resid, int ldr,
    float* __restrict__ C, int ldc, int K)
{
    __shared__ float lds_a[2][BM * S20];   // 10 KB x 2
    __shared__ float lds_b[2][BM * S20];   // 10 KB x 2 (covers both layouts)

    const int tid   = threadIdx.x;
    const int lane  = tid & 31;
    const int wave  = tid >> 5;
    const int waveM = (wave >> 1) * 64;    // 2x2 wave grid, 64x64 each
    const int waveN = (wave & 1) * 64;
    const int blockM = blockIdx.y * BM;
    const int blockN = blockIdx.x * BN;

    const unsigned ldsa0 = (unsigned)(uintptr_t)&lds_a[0][0];
    const unsigned ldsa1 = (unsigned)(uintptr_t)&lds_a[1][0];
    const unsigned ldsb0 = (unsigned)(uintptr_t)&lds_b[0][0];
    const unsigned ldsb1 = (unsigned)(uintptr_t)&lds_b[1][0];

    v8f acc[4][4];
    #pragma unroll
    for (int mi = 0; mi < 4; ++mi)
        #pragma unroll
        for (int ni = 0; ni < 4; ++ni) {
            v8f z = {0.f,0.f,0.f,0.f,0.f,0.f,0.f,0.f};
            acc[mi][ni] = z;
        }

    const int mlane = lane & 15;
    const int khalf = (lane >> 4) * 2;     // ISA: lanes 16-31 hold K+2,K+3

    // prologue: stage k0 = 0 into buffer 0
    stage_tiles<BT>(A, lda, Bm, ldb, blockM, blockN, 0, ldsa0, ldsb0, tid);
    wait_async0();
    __syncthreads();

    int p = 0;
    for (int k0 = 0; k0 < K; k0 += BK) {
        // prefetch next stage into the other buffer while we compute
        if (k0 + BK < K)
            stage_tiles<BT>(A, lda, Bm, ldb, blockM, blockN, k0 + BK,
                            p ? ldsa0 : ldsa1, p ? ldsb0 : ldsb1, tid);

        const float* la = lds_a[p];
        const float* lb = lds_b[p];
        #pragma unroll
        for (int kk = 0; kk < BK; kk += 4) {
            v2f afrag[4];
            #pragma unroll
            for (int mi = 0; mi < 4; ++mi)
                afrag[mi] = *reinterpret_cast<const v2f*>(
                    &la[(waveM + mi * 16 + mlane) * S20 + kk + khalf]);
            v2f bfrag[4];
            #pragma unroll
            for (int ni = 0; ni < 4; ++ni) {
                int ncol = waveN + ni * 16 + mlane;
                if (BT) {
                    bfrag[ni] = *reinterpret_cast<const v2f*>(
                        &lb[ncol * S20 + kk + khalf]);
                } else {
                    v2f bf;
                    bf.x = lb[(kk + khalf) * S_NN + ncol];
                    bf.y = lb[(kk + khalf + 1) * S_NN + ncol];
                    bfrag[ni] = bf;
                }
            }
            #pragma unroll
            for (int mi = 0; mi < 4; ++mi)
                #pragma unroll
                for (int ni = 0; ni < 4; ++ni)
                    acc[mi][ni] = __builtin_amdgcn_wmma_f32_16x16x4_f32(
                        false, afrag[mi], false, bfrag[ni],
                        (short)0, acc[mi][ni], false, false);
        }
        wait_async0();
        __syncthreads();
        p ^= 1;
    }

    // ---- epilogue: bias (+optional second bias) (+optional residual) ----
    const int lhalf = lane >> 4;
    #pragma unroll
    for (int ni = 0; ni < 4; ++ni) {
        int col = blockN + waveN + ni * 16 + mlane;
        float bb = 0.f;
        if (bias0) bb += bias0[col];
        if (bias1) bb += bias1[col];
        #pragma unroll
        for (int mi = 0; mi < 4; ++mi) {
            #pragma unroll
            for (int r = 0; r < 8; ++r) {
                int row = blockM + waveM + mi * 16 + r + lhalf * 8;
                float val = acc[mi][ni][r] + bb;
                if (resid) val += resid[(size_t)row * ldr + col];
                C[(size_t)row * ldc + col] = val;
            }
        }
    }
}

// ---------------------------------------------------------------------------
// Row softmax (1024 cols), scale folded in. One block per row.
// ---------------------------------------------------------------------------
__global__ __launch_bounds__(256) void softmax_kernel(float* __restrict__ S, float scale) {
    __shared__ float red[256];
    const int row = blockIdx.x;
    const int tid = threadIdx.x;
    float* srow = S + (size_t)row * B_SZ;
    float v[4];
    float mx = -INFINITY;
    #pragma unroll
    for (int i = 0; i < 4; ++i) {
        v[i] = srow[tid + i * 256] * scale;
        mx = fmaxf(mx, v[i]);
    }
    red[tid] = mx; __syncthreads();
    for (int s = 128; s > 0; s >>= 1) {
        if (tid < s) red[tid] = fmaxf(red[tid], red[tid + s]);
        __syncthreads();
    }
    mx = red[0]; __syncthreads();
    float sum = 0.f;
    #pragma unroll
    for (int i = 0; i < 4; ++i) { v[i] = __expf(v[i] - mx); sum += v[i]; }
    red[tid] = sum; __syncthreads();
    for (int s = 128; s > 0; s >>= 1) {
        if (tid < s) red[tid] += red[tid + s];
        __syncthreads();
    }
    float inv = 1.f / red[0];
    #pragma unroll
    for (int i = 0; i < 4; ++i) srow[tid + i * 256] = v[i] * inv;
}

// ---------------------------------------------------------------------------
// LSTM cell activations; out = concat(h_f, c_b), width 2H = 1024
// ---------------------------------------------------------------------------
__global__ __launch_bounds__(256) void lstm_out_kernel(const float* __restrict__ gf,
                                                       const float* __restrict__ gb,
                                                       float* __restrict__ out) {
    int idx = blockIdx.x * 256 + threadIdx.x;    // over B*H
    int b = idx / H_SZ;
    int j = idx - b * H_SZ;
    const float* gfr = gf + (size_t)b * 4 * H_SZ;
    float i_f = gfr[j], g_f = gfr[2 * H_SZ + j], o_f = gfr[3 * H_SZ + j];
    float c_f = sigm(i_f) * tanhf(g_f);
    float h_f = sigm(o_f) * tanhf(c_f);
    const float* gbr = gb + (size_t)b * 4 * H_SZ;
    float i_b = gbr[j], g_b = gbr[2 * H_SZ + j];
    float c_b = sigm(i_b) * tanhf(g_b);
    out[(size_t)b * 2 * H_SZ + j] = h_f;
    out[(size_t)b * 2 * H_SZ + H_SZ + j] = c_b;
}

// ---------------------------------------------------------------------------
extern "C" void kernel_launch(void* const* d_in, const int* in_sizes, int n_in,
                              void* d_out, int out_size, void* d_ws, size_t ws_size,
                              hipStream_t stream) {
    (void)in_sizes; (void)n_in; (void)out_size; (void)ws_size;
    const float* inputs = (const float*)d_in[0];
    const float* Wqkv_f = (const float*)d_in[1];
    const float* bqkv_f = (const float*)d_in[2];
    const float* Wo_f   = (const float*)d_in[3];
    const float* bo_f   = (const float*)d_in[4];
    const float* Wih_f  = (const float*)d_in[5];
    const float* bih_f  = (const float*)d_in[6];
    const float* bhh_f  = (const float*)d_in[7];
    const float* Wqkv_b = (const float*)d_in[8];
    const float* bqkv_b = (const float*)d_in[9];
    const float* Wo_b   = (const float*)d_in[10];
    const float* bo_b   = (const float*)d_in[11];
    const float* Wih_b  = (const float*)d_in[12];
    const float* bih_b  = (const float*)d_in[13];
    const float* bhh_b  = (const float*)d_in[14];

    float* ws = (float*)d_ws;
    const size_t M1 = (size_t)1 << 20;           // 1M floats = B*E
    float* x     = ws;
    float* xr    = ws + 1 * M1;
    float* qkv_f = ws + 2 * M1;                  // 3M
    float* qkv_b = ws + 5 * M1;                  // 3M
    float* sc_f  = ws + 8 * M1;
    float* sc_b  = ws + 9 * M1;
    float* av_f  = ws + 10 * M1;
    float* av_b  = ws + 11 * M1;
    float* af_f  = ws + 12 * M1;
    float* af_b  = ws + 13 * M1;
    float* g_f   = ws + 14 * M1;                 // 2M
    float* g_b   = ws + 16 * M1;                 // 2M

    const int E3 = 3 * E_SZ, H4 = 4 * H_SZ;
    dim3 gblk(128);
    dim3 blk(256);
    const float scale = 0.03125f;                // 1/sqrt(1024)

    prep_kernel<<<(B_SZ * E_SZ) / 256, blk, 0, stream>>>(inputs, x, xr);

    // qkv = x @ Wqkv.T + bqkv        (M=1024, N=3072, K=1024)
    gemm_wmma_kernel<true><<<dim3(E3 / BN, B_SZ / BM), gblk, 0, stream>>>(
        x, E_SZ, Wqkv_f, E_SZ, bqkv_f, nullptr, nullptr, 0, qkv_f, E3, E_SZ);
    gemm_wmma_kernel<true><<<dim3(E3 / BN, B_SZ / BM), gblk, 0, stream>>>(
        xr, E_SZ, Wqkv_b, E_SZ, bqkv_b, nullptr, nullptr, 0, qkv_b, E3, E_SZ);

    // scores = q @ k.T               (M=N=1024, K=1024)
    gemm_wmma_kernel<true><<<dim3(B_SZ / BN, B_SZ / BM), gblk, 0, stream>>>(
        qkv_f, E3, qkv_f + E_SZ, E3, nullptr, nullptr, nullptr, 0, sc_f, B_SZ, E_SZ);
    gemm_wmma_kernel<true><<<dim3(B_SZ / BN, B_SZ / BM), gblk, 0, stream>>>(
        qkv_b, E3, qkv_b + E_SZ, E3, nullptr, nullptr, nullptr, 0, sc_b, B_SZ, E_SZ);

    softmax_kernel<<<B_SZ, blk, 0, stream>>>(sc_f, scale);
    softmax_kernel<<<B_SZ, blk, 0, stream>>>(sc_b, scale);

    // av = a @ v                     (B is K x N with row stride 3E)
    gemm_wmma_kernel<false><<<dim3(E_SZ / BN, B_SZ / BM), gblk, 0, stream>>>(
        sc_f, B_SZ, qkv_f + 2 * E_SZ, E3, nullptr, nullptr, nullptr, 0, av_f, E_SZ, B_SZ);
    gemm_wmma_kernel<false><<<dim3(E_SZ / BN, B_SZ / BM), gblk, 0, stream>>>(
        sc_b, B_SZ, qkv_b + 2 * E_SZ, E3, nullptr, nullptr, nullptr, 0, av_b, E_SZ, B_SZ);

    // af = av @ Wo.T + bo + x        (residual fused in epilogue)
    gemm_wmma_kernel<true><<<dim3(E_SZ / BN, B_SZ / BM), gblk, 0, stream>>>(
        av_f, E_SZ, Wo_f, E_SZ, bo_f, nullptr, x, E_SZ, af_f, E_SZ, E_SZ);
    gemm_wmma_kernel<true><<<dim3(E_SZ / BN, B_SZ / BM), gblk, 0, stream>>>(
        av_b, E_SZ, Wo_b, E_SZ, bo_b, nullptr, xr, E_SZ, af_b, E_SZ, E_SZ);

    // gates = af @ W_ih.T + (b_ih + b_hh)    (N = 4H = 2048)
    gemm_wmma_kernel<true><<<dim3(H4 / BN, B_SZ / BM), gblk, 0, stream>>>(
        af_f, E_SZ, Wih_f, E_SZ, bih_f, bhh_f, nullptr, 0, g_f, H4, E_SZ);
    gemm_wmma_kernel<true><<<dim3(H4 / BN, B_SZ / BM), gblk, 0, stream>>>(
        af_b, E_SZ, Wih_b, E_SZ, bih_b, bhh_b, nullptr, 0, g_b, H4, E_SZ);

    lstm_out_kernel<<<(B_SZ * H_SZ) / 256, blk, 0, stream>>>(g_f, g_b, (float*)d_out);
}